// STGI_47571057770868
// MI455X (gfx1250) — compile-verified
//
#include <hip/hip_runtime.h>

// ---------------------------------------------------------------------------
// Factored 2-layer GCN (C=1):   out = where(mask, x, A·mlp(A·x))
//   A = D^-1/2 (W + I) D^-1/2  applied as scalar SpMV per replica (G=64)
//   mlp(s) = W2^T relu(W1*s + b1)   (b2 added with the final self-loop pass)
// Node-major (N,64) transposed layout => each edge touches one contiguous
// 256B block for gather and one for scatter. Sparse passes: float4 gathers +
// hardware f32 atomics (16 threads/edge). Per-node MLP: V_WMMA_F32_16X16X4_F32.
// ---------------------------------------------------------------------------

typedef float v2f __attribute__((ext_vector_type(2)));
typedef float v8f __attribute__((ext_vector_type(8)));

__global__ void k_init_deg(float* __restrict__ deg, int N) {
    int i = blockIdx.x * blockDim.x + threadIdx.x;
    if (i < N) deg[i] = 1.0f;               // self-loop weight
}

__global__ void k_accum_deg(const int* __restrict__ dst, const float* __restrict__ w,
                            float* __restrict__ deg, int E) {
    int e = blockIdx.x * blockDim.x + threadIdx.x;
    if (e < E) unsafeAtomicAdd(&deg[dst[e]], w[e]);
}

__global__ void k_dinv(float* __restrict__ deg, float* __restrict__ selfn, int N) {
    int i = blockIdx.x * blockDim.x + threadIdx.x;
    if (i < N) {
        float d = deg[i];
        float r = d > 0.0f ? rsqrtf(d) : 0.0f;
        deg[i] = r;                         // deg now holds dinv
        selfn[i] = r * r;                   // self-loop norm
    }
}

__global__ void k_norm(const int* __restrict__ src, const int* __restrict__ dst,
                       const float* __restrict__ w, const float* __restrict__ dinv,
                       float* __restrict__ norm, int E) {
    int e = blockIdx.x * blockDim.x + threadIdx.x;
    if (e < E) norm[e] = dinv[src[e]] * w[e] * dinv[dst[e]];
}

// xT[n*64+g] = x[g*N+n];  sT = selfnorm[n]*xT  (self-loop term of SpMV #1)
__global__ void k_transpose_init(const float* __restrict__ x, const float* __restrict__ selfn,
                                 float* __restrict__ xT, float* __restrict__ sT,
                                 int N, int total) {
    int i = blockIdx.x * blockDim.x + threadIdx.x;   // i = n*64 + g
    if (i < total) {
        int n = i >> 6, g = i & 63;         // G == 64
        float v = x[g * N + n];
        xT[i] = v;
        sT[i] = selfn[n] * v;
    }
}

// One edge per 16 consecutive threads: float4 (b128) gather of 4 replicas,
// 4 hardware f32 atomic scatters. Edge payloads are 256B-contiguous blocks.
__global__ void k_spmv_scatter(const int* __restrict__ src, const int* __restrict__ dst,
                               const float* __restrict__ norm,
                               const float* __restrict__ vin, float* __restrict__ vout,
                               int E) {
    int t = blockIdx.x * blockDim.x + threadIdx.x;
    int e = t >> 4;
    int g4 = (t & 15) << 2;                  // replica group: 4 floats, 16B aligned
    if (e < E) {
        if (g4 == 0) {                       // global_prefetch_b8 the edge stream
            __builtin_prefetch(src + e + 1024, 0, 0);
            __builtin_prefetch(dst + e + 1024, 0, 0);
            __builtin_prefetch(norm + e + 1024, 0, 0);
        }
        int s = src[e], d = dst[e];
        float nw = norm[e];
        const float4 v = *(const float4*)(vin + (s << 6) + g4);
        float* po = vout + (d << 6) + g4;
        unsafeAtomicAdd(po + 0, nw * v.x);
        unsafeAtomicAdd(po + 1, nw * v.y);
        unsafeAtomicAdd(po + 2, nw * v.z);
        unsafeAtomicAdd(po + 3, nw * v.w);
    }
}

// z[i] = sum_h W2[h]*relu(W1[h]*s[i]+b1[h]) for 16 items per wave via WMMA.
// A (16x4): col0 = W1 chunk (lanes 0-15, VGPR0).  B (4x16): row0 = s items.
// C carries b1 broadcast per row.  D layout: lane l -> col l%16, row v+8*(l>=16).
__global__ __launch_bounds__(256) void k_mlp_wmma(
        const float* __restrict__ sIn, float* __restrict__ zOut,
        const float* __restrict__ W1, const float* __restrict__ b1,
        const float* __restrict__ W2, int nTiles) {
    int wave = (blockIdx.x * blockDim.x + threadIdx.x) >> 5;
    int lane = threadIdx.x & 31;
    if (wave >= nTiles) return;              // wave-uniform: EXEC stays all-1s
    int base = wave * 16;
    int lo = lane & 15;
    bool loHalf = lane < 16;

    float sval = sIn[base + lo];
    v2f B;  B.x = loHalf ? sval : 0.0f;  B.y = 0.0f;

    float acc = 0.0f;
#pragma unroll
    for (int c = 0; c < 2; ++c) {            // hidden dim 32 = two 16-row chunks
        float w1v = W1[c * 16 + lo];
        v2f A;  A.x = loHalf ? w1v : 0.0f;  A.y = 0.0f;
        v8f C;
#pragma unroll
        for (int v = 0; v < 8; ++v) {
            int m = c * 16 + v + (loHalf ? 0 : 8);   // row (hidden index)
            C[v] = b1[m];
        }
        v8f D = __builtin_amdgcn_wmma_f32_16x16x4_f32(
                    false, A, false, B, (short)0, C, false, false);
#pragma unroll
        for (int v = 0; v < 8; ++v) {
            int m = c * 16 + v + (loHalf ? 0 : 8);
            float r = D[v] > 0.0f ? D[v] : 0.0f;     // relu
            acc += r * W2[m];                        // W2-weighted row reduction
        }
    }
    acc += __shfl_xor(acc, 16, 32);          // fold upper row-half into lanes 0-15
    if (lane < 16) zOut[base + lane] = acc;
}

// outT = selfnorm[n]*zT + b2  (self-loop term + bias of SpMV #2; full overwrite)
__global__ void k_init_out(const float* __restrict__ zT, const float* __restrict__ selfn,
                           const float* __restrict__ b2, float* __restrict__ outT,
                           int total4) {
    int i = blockIdx.x * blockDim.x + threadIdx.x;   // one float4 per thread
    if (i < total4) {
        float sn = selfn[i >> 4];            // (i*4)>>6: node of this 4-group
        float bb = b2[0];
        float4 z = ((const float4*)zT)[i];
        float4 o;
        o.x = sn * z.x + bb;  o.y = sn * z.y + bb;
        o.z = sn * z.z + bb;  o.w = sn * z.w + bb;
        ((float4*)outT)[i] = o;
    }
}

__global__ void k_finalize(const float* __restrict__ x, const unsigned char* __restrict__ mask,
                           const float* __restrict__ outT, float* __restrict__ out,
                           int N, int total) {
    int i = blockIdx.x * blockDim.x + threadIdx.x;   // i = g*N + n (output order)
    if (i < total) {
        int g = i / N, n = i - g * N;
        out[i] = mask[i] ? x[i] : outT[(n << 6) + g];
    }
}

extern "C" void kernel_launch(void* const* d_in, const int* in_sizes, int n_in,
                              void* d_out, int out_size, void* d_ws, size_t ws_size,
                              hipStream_t stream) {
    const float*         x    = (const float*)d_in[0];
    const unsigned char* mask = (const unsigned char*)d_in[1];   // numpy bool (1B)
    const int*           ei   = (const int*)d_in[2];
    const float*         ew   = (const float*)d_in[3];
    const float*         W1   = (const float*)d_in[4];
    const float*         b1   = (const float*)d_in[5];
    const float*         W2   = (const float*)d_in[6];
    const float*         b2   = (const float*)d_in[7];
    float*               out  = (float*)d_out;

    const int N     = 10000;                 // N_NODES (reference constant)
    const int total = in_sizes[0];           // B*S*N*C = N*G, G = 64
    const int E     = in_sizes[3];           // 160000

    const int* src = ei;
    const int* dst = ei + E;

    // Workspace layout (floats); bufA: xT then zT, bufB: sT then outT.
    float* ws = (float*)d_ws;
    size_t o = 0;
    float* deg   = ws + o;  o += (size_t)((N + 255) & ~255);
    float* selfn = ws + o;  o += (size_t)((N + 255) & ~255);
    float* norm  = ws + o;  o += (size_t)((E + 255) & ~255);
    float* bufA  = ws + o;  o += (size_t)total;
    float* bufB  = ws + o;  o += (size_t)total;
    (void)ws_size; (void)n_in; (void)out_size;

    dim3 blk(256);
    k_init_deg      <<<(N + 255) / 256, blk, 0, stream>>>(deg, N);
    k_accum_deg     <<<(E + 255) / 256, blk, 0, stream>>>(dst, ew, deg, E);
    k_dinv          <<<(N + 255) / 256, blk, 0, stream>>>(deg, selfn, N);
    k_norm          <<<(E + 255) / 256, blk, 0, stream>>>(src, dst, ew, deg, norm, E);
    k_transpose_init<<<(total + 255) / 256, blk, 0, stream>>>(x, selfn, bufA, bufB, N, total);

    long scatterThreads = (long)E * 16;      // 16 threads per edge (float4 lanes)
    k_spmv_scatter  <<<(unsigned)((scatterThreads + 255) / 256), blk, 0, stream>>>(
                        src, dst, norm, bufA, bufB, E);          // sT += A-edges * xT

    int nTiles = total / 16;                 // 40000 tiles, one per wave
    k_mlp_wmma      <<<(nTiles * 32 + 255) / 256, blk, 0, stream>>>(
                        bufB, bufA, W1, b1, W2, nTiles);         // zT = mlp(sT)

    int total4 = total / 4;
    k_init_out      <<<(total4 + 255) / 256, blk, 0, stream>>>(bufA, selfn, b2, bufB, total4);
    k_spmv_scatter  <<<(unsigned)((scatterThreads + 255) / 256), blk, 0, stream>>>(
                        src, dst, norm, bufA, bufB, E);          // outT += A-edges * zT

    k_finalize      <<<(total + 255) / 256, blk, 0, stream>>>(x, mask, bufB, out, N, total);
}